// ImprovedSupervisedContrastiveLoss_7146825581127
// MI455X (gfx1250) — compile-verified
//
#include <hip/hip_runtime.h>
#include <hip/hip_bf16.h>

typedef float v2f __attribute__((ext_vector_type(2)));
typedef float v8f __attribute__((ext_vector_type(8)));

#define NUM_CLASSES 20
#define D 64
#define INV_T (1.0f / 0.07f)

// ---------------------------------------------------------------------------
// K0: zero the vote buffer
// ---------------------------------------------------------------------------
__global__ void k_zero_i32(int* p, int n) {
    int i = blockIdx.x * blockDim.x + threadIdx.x;
    if (i < n) p[i] = 0;
}

// ---------------------------------------------------------------------------
// K1: vote raw-point labels into superpoints (integer atomics: deterministic)
// ---------------------------------------------------------------------------
__global__ void k_vote(const int* __restrict__ r2s, const int* __restrict__ lab,
                       int* __restrict__ votes, int n_raw) {
    int i = blockIdx.x * blockDim.x + threadIdx.x;
    if (i < n_raw)
        atomicAdd(&votes[r2s[i] * NUM_CLASSES + lab[i]], 1);
}

// ---------------------------------------------------------------------------
// K2: argmax (first max wins, like jnp.argmax) -> superpoint labels
// ---------------------------------------------------------------------------
__global__ void k_splabel(const int* __restrict__ votes, int* __restrict__ labels,
                          int num_sp, int n_raw) {
    int s = blockIdx.x * blockDim.x + threadIdx.x;
    if (s >= num_sp) return;
    int best = 0, bc = votes[s * NUM_CLASSES];
    for (int c = 1; c < NUM_CLASSES; ++c) {
        int v = votes[s * NUM_CLASSES + c];
        if (v > bc) { bc = v; best = c; }
    }
    labels[n_raw + s] = best;
}

// ---------------------------------------------------------------------------
// K3: concat + L2 normalize rows into padded buffer; pad rows = 0, label -1
// ---------------------------------------------------------------------------
__global__ void k_normalize(const float* __restrict__ rawF, const float* __restrict__ spF,
                            const int* __restrict__ lab, float* __restrict__ allF,
                            int* __restrict__ labels, int n_raw, int ntot, int npad) {
    int i = blockIdx.x * blockDim.x + threadIdx.x;
    if (i >= npad) return;
    float* dst = allF + (size_t)i * D;
    if (i >= ntot) {
        for (int d = 0; d < D; ++d) dst[d] = 0.0f;
        labels[i] = -1;
        return;
    }
    const float* src = (i < n_raw) ? (rawF + (size_t)i * D)
                                   : (spF + (size_t)(i - n_raw) * D);
    float nsq = 0.0f;
    for (int d = 0; d < D; ++d) { float x = src[d]; nsq += x * x; }
    float m = fmaxf(sqrtf(nsq), 1e-12f);
    for (int d = 0; d < D; ++d) dst[d] = src[d] / m;
    if (i < n_raw) labels[i] = lab[i];
}

// ---------------------------------------------------------------------------
// K4: per-class feature sums S_c [20 x 64] + histogram (deterministic scan)
// ---------------------------------------------------------------------------
__global__ void k_classsum(const float* __restrict__ allF, const int* __restrict__ labels,
                           float* __restrict__ S, int* __restrict__ hist, int ntot) {
    int c = blockIdx.x;
    int d = threadIdx.x;
    float s = 0.0f;
    int cnt = 0;
    for (int i = 0; i < ntot; ++i) {
        if (labels[i] == c) {
            s += allF[(size_t)i * D + d];
            if (d == 0) ++cnt;
        }
    }
    S[c * D + d] = s;
    if (d == 0) hist[c] = cnt;
}

// ---------------------------------------------------------------------------
// K5: denom_i = sum_{j != i, j < ntot} exp(f_i . f_j / T)
// One wave per 16-row tile. 4 waves/block share each 16x64 column tile in LDS.
// Gram via V_WMMA_F32_16X16X4_F32, K accumulated over 16 steps (K=64).
// ---------------------------------------------------------------------------
__global__ void __launch_bounds__(128)
k_denom(const float* __restrict__ allF, float* __restrict__ denom,
        int ntot, int ntiles) {
    __shared__ float tileB[16 * D];   // 4 KB column tile shared by 4 waves

    const int tid  = threadIdx.x;
    const int wave = tid >> 5;
    const int lane = tid & 31;
    const int lr   = lane & 15;   // A-row / B-col / C-col within tile
    const int hi   = lane >> 4;   // half-wave selector

    int rt = blockIdx.x * 4 + wave;
    bool valid = rt < ntiles;
    int rtc = valid ? rt : (ntiles - 1);

    // A fragments: 16 K-chunks, each v2f.  K = 4*k + 2*hi + {0,1}, M = lr.
    v2f a[16];
    {
        const float* arow = allF + (size_t)(rtc * 16 + lr) * D + 2 * hi;
        #pragma unroll
        for (int k = 0; k < 16; ++k)
            a[k] = *(const v2f*)(arow + 4 * k);
    }

    float acc[8];
    #pragma unroll
    for (int v = 0; v < 8; ++v) acc[v] = 0.0f;

    const int growbase = rtc * 16 + hi * 8;

    for (int ct = 0; ct < ntiles; ++ct) {
        // cooperative load of column tile: 1024 floats = 256 float4, 128 threads
        __syncthreads();
        {
            const float4* src = (const float4*)(allF + (size_t)ct * 16 * D);
            float4* dstv = (float4*)tileB;
            dstv[tid]       = src[tid];
            dstv[tid + 128] = src[tid + 128];
        }
        __syncthreads();

        // B fragments from LDS: same addressing as A (Gram symmetry)
        v8f c = {};
        #pragma unroll
        for (int k = 0; k < 16; ++k) {
            v2f b = *(const v2f*)(&tileB[lr * D + 4 * k + 2 * hi]);
            c = __builtin_amdgcn_wmma_f32_16x16x4_f32(
                    false, a[k], false, b, (short)0, c, false, false);
        }

        const int gcol = ct * 16 + lr;
        if (gcol < ntot) {
            #pragma unroll
            for (int v = 0; v < 8; ++v) {
                int grow = growbase + v;
                if (gcol != grow)
                    acc[v] += __expf(c[v] * INV_T);
            }
        }
    }

    // reduce each row partial across its 16-lane group (wave32)
    #pragma unroll
    for (int v = 0; v < 8; ++v) {
        float s = acc[v];
        s += __shfl_xor(s, 1, 16);
        s += __shfl_xor(s, 2, 16);
        s += __shfl_xor(s, 4, 16);
        s += __shfl_xor(s, 8, 16);
        acc[v] = s;
    }
    if (valid && lr == 0) {
        int base = rt * 16 + hi * 8;
        #pragma unroll
        for (int v = 0; v < 8; ++v) {
            int row = base + v;
            if (row < ntot) denom[row] = acc[v];
        }
    }
}

// ---------------------------------------------------------------------------
// K6: finalize loss (single block, deterministic tree reduction)
// ---------------------------------------------------------------------------
__global__ void k_finalize(const float* __restrict__ allF, const int* __restrict__ labels,
                           const float* __restrict__ denom, const float* __restrict__ S,
                           const int* __restrict__ hist, float* __restrict__ out, int ntot) {
    __shared__ float red[256];
    float sum = 0.0f;
    for (int i = threadIdx.x; i < ntot; i += blockDim.x) {
        const float* f  = allF + (size_t)i * D;
        int c = labels[i];
        const float* sc = S + c * D;
        float selfdot = 0.0f, posdot = 0.0f;
        for (int d = 0; d < D; ++d) {
            float fd = f[d];
            selfdot += fd * fd;
            posdot  += fd * sc[d];
        }
        float cnt  = (float)(hist[c] - 1);
        float logd = logf(denom[i] + 1e-9f);
        float mean = ((posdot - selfdot) * INV_T - cnt * logd) / (cnt + 1e-9f);
        sum += mean;
    }
    red[threadIdx.x] = sum;
    __syncthreads();
    for (int s = blockDim.x / 2; s > 0; s >>= 1) {
        if (threadIdx.x < (unsigned)s) red[threadIdx.x] += red[threadIdx.x + s];
        __syncthreads();
    }
    if (threadIdx.x == 0)
        out[0] = -red[0] / (float)ntot;   // LOSS_WEIGHT == 1
}

// ---------------------------------------------------------------------------
extern "C" void kernel_launch(void* const* d_in, const int* in_sizes, int n_in,
                              void* d_out, int out_size, void* d_ws, size_t ws_size,
                              hipStream_t stream) {
    const float* spF  = (const float*)d_in[0];   // [num_sp, 64]
    const float* rawF = (const float*)d_in[1];   // [n_raw, 64]
    const int*   r2s  = (const int*)d_in[2];     // [n_raw]
    const int*   lab  = (const int*)d_in[3];     // [n_raw]
    float* out = (float*)d_out;

    const int num_sp = in_sizes[0] / D;
    const int n_raw  = in_sizes[2];
    const int ntot   = n_raw + num_sp;
    const int npad   = (ntot + 15) & ~15;
    const int ntiles = npad / 16;

    // workspace carve-up (~2.7 MB total)
    float* allF   = (float*)d_ws;                          // npad*64
    int*   labels = (int*)(allF + (size_t)npad * D);       // npad
    float* denom  = (float*)(labels + npad);               // npad
    float* S      = denom + npad;                          // 20*64
    int*   hist   = (int*)(S + NUM_CLASSES * D);           // 20
    int*   votes  = hist + NUM_CLASSES;                    // num_sp*20

    const int nv = num_sp * NUM_CLASSES;
    k_zero_i32<<<(nv + 255) / 256, 256, 0, stream>>>(votes, nv);
    k_vote<<<(n_raw + 255) / 256, 256, 0, stream>>>(r2s, lab, votes, n_raw);
    k_splabel<<<(num_sp + 255) / 256, 256, 0, stream>>>(votes, labels, num_sp, n_raw);
    k_normalize<<<(npad + 255) / 256, 256, 0, stream>>>(rawF, spF, lab, allF, labels,
                                                        n_raw, ntot, npad);
    k_classsum<<<NUM_CLASSES, D, 0, stream>>>(allF, labels, S, hist, ntot);
    k_denom<<<(ntiles + 3) / 4, 128, 0, stream>>>(allF, denom, ntot, ntiles);
    k_finalize<<<1, 256, 0, stream>>>(allF, labels, denom, S, hist, out, ntot);
}